// Net_63496796504125
// MI455X (gfx1250) — compile-verified
//
#include <hip/hip_runtime.h>
#include <hip/hip_bf16.h>

typedef __attribute__((ext_vector_type(2))) float v2f;
typedef __attribute__((ext_vector_type(8))) float v8f;

#define ALPHA 0.1f
#define HD 8
#define F_IN 64

// ---------- utility fills ----------
__global__ void k_fill(float* __restrict__ p, int n, float v) {
    int i = blockIdx.x * blockDim.x + threadIdx.x;
    if (i < n) p[i] = v;
}

// ---------- degree accumulation: deg[col[e]] += w[e]  (deg pre-filled with 1.0 = self loop) ----------
__global__ void k_deg(const int* __restrict__ col, const float* __restrict__ w,
                      float* __restrict__ deg, int E) {
    int e = blockIdx.x * blockDim.x + threadIdx.x;
    if (e < E) atomicAdd(&deg[col[e]], w[e]);
}

// ---------- deg -> dinv = rsqrt(deg) in place ----------
__global__ void k_rsqrt(float* __restrict__ d, int n) {
    int i = blockIdx.x * blockDim.x + threadIdx.x;
    if (i < n) { float v = d[i]; d[i] = (v > 0.f) ? rsqrtf(v) : 0.f; }
}

// ---------- y = x @ W1 via V_WMMA_F32_16X16X4_F32; one wave -> 16 rows x 8 (padded 16) cols ----
// Branch-free fragment loads: clamp out-of-range row/col indices instead of predicating;
// clamped rows/cols only affect output rows/cols that are never stored.
__global__ void k_gemm_xW1(const float* __restrict__ x, const float* __restrict__ W1,
                           float* __restrict__ y, int N) {
    int gtid = blockIdx.x * blockDim.x + threadIdx.x;
    int wave = gtid >> 5;
    int lane = threadIdx.x & 31;
    int half = lane >> 4;      // 0: lanes 0-15, 1: lanes 16-31
    int ln   = lane & 15;
    int m0 = wave * 16;
    if (m0 >= N) return;

    int row  = m0 + ln;
    int rowc = (row < N) ? row : (N - 1);      // clamp (row >= N never stored)
    int lnc  = (ln < HD) ? ln : (HD - 1);      // clamp (col >= 8 never stored)

    // A fragment: 16x4 f32. lanes0-15: VGPR0=K0,VGPR1=K1; lanes16-31: VGPR0=K2,VGPR1=K3
    const float* xr = x + (size_t)rowc * F_IN + 2 * half;   // 8B aligned
    // B fragment: 4x16 f32. VGPR0: row K=half; VGPR1: row K=2+half; col N=lane%16
    const float* wb = W1 + (size_t)half * HD + lnc;

    v8f c = {};
    #pragma unroll
    for (int k = 0; k < F_IN; k += 4) {
        v2f a = *(const v2f*)(xr + k);          // one global_load_b64, no branch
        v2f b;
        b.x = wb[(size_t)k * HD];
        b.y = wb[(size_t)(k + 2) * HD];
        c = __builtin_amdgcn_wmma_f32_16x16x4_f32(false, a, false, b, (short)0, c, false, false);
    }
    // C/D layout: VGPR v, lanes 0-15 -> M=v, lanes 16-31 -> M=v+8; N=lane%16
    if (ln < HD) {
        if (m0 + 16 <= N) {
            #pragma unroll
            for (int v = 0; v < 8; ++v)
                y[(size_t)(m0 + v + 8 * half) * HD + ln] = c[v];
        } else {
            #pragma unroll
            for (int v = 0; v < 8; ++v) {
                int m = m0 + v + 8 * half;
                if (m < N) y[(size_t)m * HD + ln] = c[v];
            }
        }
    }
}

// ---------- edge propagation in H=8 space: dst[col] += dinv[row]*w*dinv[col] * src[row] ----------
__global__ void k_prop(const int* __restrict__ ei, const float* __restrict__ w,
                       const float* __restrict__ dinv, const float* __restrict__ src,
                       float* __restrict__ dst, int E) {
    int e = blockIdx.x * blockDim.x + threadIdx.x;
    if (e >= E) return;
    int r = ei[e];
    int c = ei[E + e];
    float nrm = dinv[r] * w[e] * dinv[c];
    const float4* s = (const float4*)(src + (size_t)r * HD);
    float4 lo = s[0], hi = s[1];
    float* o = dst + (size_t)c * HD;
    atomicAdd(o + 0, nrm * lo.x); atomicAdd(o + 1, nrm * lo.y);
    atomicAdd(o + 2, nrm * lo.z); atomicAdd(o + 3, nrm * lo.w);
    atomicAdd(o + 4, nrm * hi.x); atomicAdd(o + 5, nrm * hi.y);
    atomicAdd(o + 6, nrm * hi.z); atomicAdd(o + 7, nrm * hi.w);
}

// ---------- h = relu(alpha*y + (1-alpha)*(agg + dinv^2*y) + b1)  (self-loop folded in) ----------
__global__ void k_combine1(const float* __restrict__ y, const float* __restrict__ agg,
                           const float* __restrict__ dinv, const float* __restrict__ b1,
                           float* __restrict__ h, int N) {
    int i = blockIdx.x * blockDim.x + threadIdx.x;
    if (i >= N) return;
    float d2 = dinv[i] * dinv[i];
    const float4* y4 = (const float4*)(y + (size_t)i * HD);
    const float4* a4 = (const float4*)(agg + (size_t)i * HD);
    const float4* b4 = (const float4*)b1;
    float4* h4 = (float4*)(h + (size_t)i * HD);
    #pragma unroll
    for (int q = 0; q < 2; ++q) {
        float4 yv = y4[q], av = a4[q], bv = b4[q];
        float4 r;
        r.x = fmaxf(ALPHA * yv.x + (1.f - ALPHA) * (av.x + d2 * yv.x) + bv.x, 0.f);
        r.y = fmaxf(ALPHA * yv.y + (1.f - ALPHA) * (av.y + d2 * yv.y) + bv.y, 0.f);
        r.z = fmaxf(ALPHA * yv.z + (1.f - ALPHA) * (av.z + d2 * yv.z) + bv.z, 0.f);
        r.w = fmaxf(ALPHA * yv.w + (1.f - ALPHA) * (av.w + d2 * yv.w) + bv.w, 0.f);
        h4[q] = r;
    }
}

// ---------- layer2 combine + 8x8 GEMM + ReLU + sum-pool (wave32 shfl reduction + atomics) ----------
__global__ void k_combine2_pool(const float* __restrict__ h1, const float* __restrict__ agg,
                                const float* __restrict__ dinv, const float* __restrict__ W2,
                                const float* __restrict__ b2, float* __restrict__ g, int N) {
    __shared__ float sW[HD * HD];
    __shared__ float sb[HD];
    if (threadIdx.x < HD * HD) sW[threadIdx.x] = W2[threadIdx.x];
    if (threadIdx.x < HD)      sb[threadIdx.x] = b2[threadIdx.x];
    __syncthreads();

    int i = blockIdx.x * blockDim.x + threadIdx.x;
    float h2[HD];
    if (i < N) {
        float d2 = dinv[i] * dinv[i];
        const float4* x4 = (const float4*)(h1 + (size_t)i * HD);
        const float4* a4 = (const float4*)(agg + (size_t)i * HD);
        float z[HD];
        #pragma unroll
        for (int q = 0; q < 2; ++q) {
            float4 hv = x4[q], av = a4[q];
            z[4 * q + 0] = ALPHA * hv.x + (1.f - ALPHA) * (av.x + d2 * hv.x);
            z[4 * q + 1] = ALPHA * hv.y + (1.f - ALPHA) * (av.y + d2 * hv.y);
            z[4 * q + 2] = ALPHA * hv.z + (1.f - ALPHA) * (av.z + d2 * hv.z);
            z[4 * q + 3] = ALPHA * hv.w + (1.f - ALPHA) * (av.w + d2 * hv.w);
        }
        #pragma unroll
        for (int k = 0; k < HD; ++k) {
            float acc = sb[k];
            #pragma unroll
            for (int j = 0; j < HD; ++j) acc += z[j] * sW[j * HD + k];
            h2[k] = fmaxf(acc, 0.f);
        }
    } else {
        #pragma unroll
        for (int k = 0; k < HD; ++k) h2[k] = 0.f;
    }
    // wave32 reduction per channel
    #pragma unroll
    for (int off = 16; off > 0; off >>= 1) {
        #pragma unroll
        for (int k = 0; k < HD; ++k) h2[k] += __shfl_xor(h2[k], off, 32);
    }
    if ((threadIdx.x & 31) == 0) {
        #pragma unroll
        for (int k = 0; k < HD; ++k) atomicAdd(&g[k], h2[k]);
    }
}

// ---------- scalar MLP head: out = relu(g@Wl1+bl1) @ Wl2 + bl2 ----------
__global__ void k_head(const float* __restrict__ g, const float* __restrict__ Wl1,
                       const float* __restrict__ bl1, const float* __restrict__ Wl2,
                       const float* __restrict__ bl2, float* __restrict__ out) {
    if (threadIdx.x != 0 || blockIdx.x != 0) return;
    float t[4];
    #pragma unroll
    for (int j = 0; j < 4; ++j) {
        float a = bl1[j];
        #pragma unroll
        for (int k = 0; k < HD; ++k) a += g[k] * Wl1[k * 4 + j];
        t[j] = fmaxf(a, 0.f);
    }
    float o = bl2[0];
    #pragma unroll
    for (int j = 0; j < 4; ++j) o += t[j] * Wl2[j];
    out[0] = o;
}

extern "C" void kernel_launch(void* const* d_in, const int* in_sizes, int n_in,
                              void* d_out, int out_size, void* d_ws, size_t ws_size,
                              hipStream_t stream) {
    const float* x   = (const float*)d_in[0];
    const int*   ei  = (const int*)d_in[1];   // [2, E] flat
    const float* ea  = (const float*)d_in[2]; // [E]
    const float* W1  = (const float*)d_in[3];
    const float* b1  = (const float*)d_in[4];
    const float* W2  = (const float*)d_in[5];
    const float* b2  = (const float*)d_in[6];
    const float* Wl1 = (const float*)d_in[7];
    const float* bl1 = (const float*)d_in[8];
    const float* Wl2 = (const float*)d_in[9];
    const float* bl2 = (const float*)d_in[10];
    float* out = (float*)d_out;

    const int N = in_sizes[0] / F_IN;
    const int E = in_sizes[2];

    // workspace layout (floats):
    float* W    = (float*)d_ws;
    float* dinv = W;                   // [N]   (degree, then rsqrt in place)
    float* y1   = W + (size_t)N;       // [N*8]
    float* h1   = W + (size_t)9 * N;   // [N*8]
    float* agg1 = W + (size_t)17 * N;  // [N*8]
    float* agg2 = W + (size_t)25 * N;  // [N*8]
    float* g    = W + (size_t)33 * N;  // [8]

    const int B = 256;
    int nzero = 16 * N + 8;  // agg1, agg2, g are contiguous

    k_fill<<<(N + B - 1) / B, B, 0, stream>>>(dinv, N, 1.0f);            // self-loop weight
    k_fill<<<(nzero + B - 1) / B, B, 0, stream>>>(agg1, nzero, 0.0f);

    k_deg<<<(E + B - 1) / B, B, 0, stream>>>(ei + E, ea, dinv, E);
    k_rsqrt<<<(N + B - 1) / B, B, 0, stream>>>(dinv, N);

    // Layer 1: fold W1 before propagation (A(xW) == (Ax)W) -> propagate in H=8 space
    int waves = (N + 15) / 16;
    int gthreads = waves * 32;
    k_gemm_xW1<<<(gthreads + B - 1) / B, B, 0, stream>>>(x, W1, y1, N);
    k_prop<<<(E + B - 1) / B, B, 0, stream>>>(ei, ea, dinv, y1, agg1, E);
    k_combine1<<<(N + B - 1) / B, B, 0, stream>>>(y1, agg1, dinv, b1, h1, N);

    // Layer 2 propagation + fused combine/GEMM/ReLU/sum-pool
    k_prop<<<(E + B - 1) / B, B, 0, stream>>>(ei, ea, dinv, h1, agg2, E);
    k_combine2_pool<<<(N + B - 1) / B, B, 0, stream>>>(h1, agg2, dinv, W2, b2, g, N);

    k_head<<<1, 1, 0, stream>>>(g, Wl1, bl1, Wl2, bl2, out);
}